// FORCEModel_69183333204549
// MI455X (gfx1250) — compile-verified
//
#include <hip/hip_runtime.h>

typedef __attribute__((ext_vector_type(2))) float v2f;
typedef __attribute__((ext_vector_type(8))) float v8f;

static constexpr int FN  = 192;   // N (state / out dim)
static constexpr int FIN = 64;    // input dim
static constexpr int FT  = 32;    // timesteps
static constexpr float FDT = 0.1f;

// ---------------------------------------------------------------------------
// Kernel 1: the FORCE cell + RLS update of P_output / W_out (single block).
//   a' = (1-dt)*a + dt*(tanh(a)@W_r + x@W_in);  h = tanh(a')
//   z = h@W_o;  e = z - y
//   k = P_o@h (symmetric -> coalesced column dot);  c = 1/(1+h.k)
//   P_o -= c k k^T          (exact: h@P_o == k by symmetry)
//   W_o -= c k e^T          (exact: P_o_new@h == c*k)
// ---------------------------------------------------------------------------
__global__ __launch_bounds__(FN) void force_cell_kernel(
    const float* __restrict__ x_t, const float* __restrict__ y_t,
    const float* __restrict__ W_in,
    float* __restrict__ a, float* __restrict__ h_out, float* __restrict__ e_out,
    float* __restrict__ P_o, float* __restrict__ W_o,
    const float* __restrict__ W_r, float* __restrict__ z_out)
{
    __shared__ float r_s[FN], x_s[FIN], h_s[FN], e_s[FN], k_s[FN], prod_s[FN];
    const int j = threadIdx.x;

    r_s[j] = tanhf(a[j]);
    if (j < FIN) x_s[j] = x_t[j];
    __syncthreads();

    float acc = 0.f;
    for (int k = 0; k < FN; ++k)  acc += r_s[k] * W_r[k * FN + j];   // coalesced cols
    for (int k = 0; k < FIN; ++k) acc += x_s[k] * W_in[k * FN + j];
    const float anew = (1.0f - FDT) * a[j] + FDT * acc;
    a[j] = anew;
    const float hh = tanhf(anew);
    h_s[j] = hh; h_out[j] = hh;
    __syncthreads();

    float z = 0.f;
    for (int k = 0; k < FN; ++k) z += h_s[k] * W_o[k * FN + j];
    const float ee = z - y_t[j];
    e_s[j] = ee; e_out[j] = ee; z_out[j] = z;

    // k = P_o @ h via columns (P_o symmetric) -> coalesced
    float kk = 0.f;
    for (int m = 0; m < FN; ++m) kk += P_o[m * FN + j] * h_s[m];
    k_s[j] = kk;
    prod_s[j] = kk * h_s[j];
    __syncthreads();

    float hk = 0.f;
    for (int m = 0; m < FN; ++m) hk += prod_s[m];   // broadcast LDS reads
    const float c = 1.0f / (1.0f + hk);

    const float ck = c * k_s[j];
    const float ce = c * e_s[j];
    for (int m = 0; m < FN; ++m) {
        const float km = k_s[m];
        P_o[m * FN + j] -= ck * km;   // P_o -= c k k^T
        W_o[m * FN + j] -= ce * km;   // W_o -= c k e^T
    }
}

// ---------------------------------------------------------------------------
// Kernel 2: per-slice RLS update of P_GG (one block per slice i) + W_r column.
//   Ph   = P_i @ h            (wave-parallel b64-vectorized row dots)
//   s    = 1/(1 + h.Ph)
//   W_r[:, i] -= s * Ph * e   (exact: Ph2 = Ph / (1+hPh) by symmetry)
//   P_i -= s * Ph Ph^T        (rank-1, 16x16 WMMA tile updates:
//                              D = A(16x4: col0 = -s*Ph) x B(4x16: row0 = Ph^T) + C)
// ---------------------------------------------------------------------------
__global__ __launch_bounds__(256) void force_pgg_kernel(
    const float* __restrict__ h_v, const float* __restrict__ e_v,
    float* __restrict__ P_g, float* __restrict__ W_r)
{
    __shared__ float h_s[FN], e_s[FN], Ph_s[FN];
    const int tid  = threadIdx.x;
    const int lane = tid & 31;
    const int wv   = tid >> 5;                  // 8 waves
    float* P = P_g + (size_t)blockIdx.x * (FN * FN);

    if (tid < FN) { h_s[tid] = h_v[tid]; e_s[tid] = e_v[tid]; }
    __syncthreads();

    // --- Phase 1: Ph = P @ h, one row per (wave, iter), float2 (b64) loads ---
    for (int n = 0; n < FN / 8; ++n) {
        const int row = wv + 8 * n;
        const v2f* rp2 = (const v2f*)(P + row * FN);
        __builtin_prefetch(P + (row + 8) * FN, 0, 3);   // next row for this wave
        float acc = 0.f;
        #pragma unroll
        for (int m = 0; m < FN / 64; ++m) {             // 3 iters of 32 lanes x f2
            const int kk = lane + 32 * m;               // float2 index, 0..95
            const v2f pv = rp2[kk];
            acc += pv.x * h_s[2 * kk] + pv.y * h_s[2 * kk + 1];
        }
        #pragma unroll
        for (int off = 16; off >= 1; off >>= 1)
            acc += __shfl_xor(acc, off, 32);
        if (lane == 0) Ph_s[row] = acc;
    }
    __syncthreads();

    float hPh = 0.f;
    for (int m = 0; m < FN; ++m) hPh += Ph_s[m] * h_s[m];
    const float s = 1.0f / (1.0f + hPh);

    // --- W_r column update (uses Ph2 = s*Ph identity; no extra P pass) ---
    if (tid < FN)
        W_r[tid * FN + blockIdx.x] -= s * Ph_s[tid] * e_s[tid];

    // --- Phase 2: P -= s * Ph Ph^T via f32 WMMA 16x16x4, 144 tiles / slice ---
    // Lane-mask multiply instead of divergent selects: lanes 0-15 carry K=0,
    // lanes 16-31 (K=2,3) carry zeros. A.y/B.y (K=1/K=3) stay zero.
    const int   nIdx  = lane & 15;
    const int   hi    = lane >> 4;
    const float mask  = (hi == 0) ? 1.0f : 0.0f;        // one v_cndmask, hoisted
    const float aScl  = -s * mask;

    int tm = 0, tn = wv;                                 // stride-8 walk over 12x12 tiles
    #pragma unroll 1
    for (int k = 0; k < (FN / 16) * (FN / 16) / 8; ++k) {
        v2f A, B;
        A.x = aScl * Ph_s[tm * 16 + nIdx];               // unconditional ds_load
        A.y = 0.f;
        B.x = mask * Ph_s[tn * 16 + nIdx];
        B.y = 0.f;

        v8f C;
        #pragma unroll
        for (int v = 0; v < 8; ++v) {
            const int row = tm * 16 + v + 8 * hi;        // C/D lane layout
            C[v] = P[row * FN + tn * 16 + nIdx];
        }
        v8f D = __builtin_amdgcn_wmma_f32_16x16x4_f32(
            /*neg_a=*/false, A, /*neg_b=*/false, B,
            /*c_mod=*/(short)0, C, /*reuse_a=*/false, /*reuse_b=*/false);
        #pragma unroll
        for (int v = 0; v < 8; ++v) {
            const int row = tm * 16 + v + 8 * hi;
            P[row * FN + tn * 16 + nIdx] = D[v];
        }

        tn += 8;                                         // no div/mod in the loop
        if (tn >= FN / 16) { tn -= FN / 16; ++tm; }
    }
}

// ---------------------------------------------------------------------------
extern "C" void kernel_launch(void* const* d_in, const int* in_sizes, int n_in,
                              void* d_out, int out_size, void* d_ws, size_t ws_size,
                              hipStream_t stream)
{
    (void)in_sizes; (void)n_in; (void)out_size; (void)ws_size;
    const float* x     = (const float*)d_in[0];  // (1,T,IN)
    const float* y     = (const float*)d_in[1];  // (1,T,N)
    const float* W_in  = (const float*)d_in[2];  // (IN,N)
    const float* W_rec = (const float*)d_in[3];  // (N,N)
    const float* W_out = (const float*)d_in[4];  // (N,N)
    const float* a0    = (const float*)d_in[5];  // (1,N)
    const float* P_out = (const float*)d_in[6];  // (N,N)
    const float* P_GG  = (const float*)d_in[7];  // (N,N,N)
    float* out = (float*)d_out;                  // (1,T,N)

    // Workspace layout (state copies; inputs stay pristine -> deterministic replays)
    float* ws = (float*)d_ws;
    float* Pg = ws;                                   // N^3
    float* Po = Pg + (size_t)FN * FN * FN;            // N^2
    float* Wo = Po + FN * FN;                         // N^2
    float* Wr = Wo + FN * FN;                         // N^2
    float* a  = Wr + FN * FN;                         // N
    float* h  = a + FN;                               // N
    float* e  = h + FN;                               // N

    hipMemcpyAsync(Pg, P_GG,  (size_t)FN * FN * FN * sizeof(float), hipMemcpyDeviceToDevice, stream);
    hipMemcpyAsync(Po, P_out, (size_t)FN * FN * sizeof(float),      hipMemcpyDeviceToDevice, stream);
    hipMemcpyAsync(Wo, W_out, (size_t)FN * FN * sizeof(float),      hipMemcpyDeviceToDevice, stream);
    hipMemcpyAsync(Wr, W_rec, (size_t)FN * FN * sizeof(float),      hipMemcpyDeviceToDevice, stream);
    hipMemcpyAsync(a,  a0,    (size_t)FN * sizeof(float),           hipMemcpyDeviceToDevice, stream);

    for (int t = 0; t < FT; ++t) {
        force_cell_kernel<<<1, FN, 0, stream>>>(
            x + t * FIN, y + t * FN, W_in, a, h, e, Po, Wo, Wr, out + t * FN);
        force_pgg_kernel<<<FN, 256, 0, stream>>>(h, e, Pg, Wr);
    }
}